// GCNZinc_78245714198776
// MI455X (gfx1250) — compile-verified
//
#include <hip/hip_runtime.h>
#include <math.h>

typedef float v2f __attribute__((ext_vector_type(2)));
typedef float v4f __attribute__((ext_vector_type(4)));
typedef float v8f __attribute__((ext_vector_type(8)));

#define DIM 128

// ---------------------------------------------------------------------------
// Degree accumulation: deg_out[src]++ , deg_in[dst]++
// ---------------------------------------------------------------------------
__global__ void degree_kernel(const int* __restrict__ src, const int* __restrict__ dst,
                              float* __restrict__ dout, float* __restrict__ din, int E) {
    int e = blockIdx.x * blockDim.x + threadIdx.x;
    if (e < E) {
        atomicAdd(dout + src[e], 1.0f);
        atomicAdd(din  + dst[e], 1.0f);
    }
}

// ns/nd = rsqrt(max(deg,1)) in place
__global__ void invsqrt_kernel(float* __restrict__ a, float* __restrict__ b, int N) {
    int n = blockIdx.x * blockDim.x + threadIdx.x;
    if (n < N) {
        a[n] = rsqrtf(fmaxf(a[n], 1.0f));
        b[n] = rsqrtf(fmaxf(b[n], 1.0f));
    }
}

// ---------------------------------------------------------------------------
// h[n] = embed_table[x[n]]   (float4 per thread)
// ---------------------------------------------------------------------------
__global__ void embed_kernel(const int* __restrict__ x, const float* __restrict__ et,
                             float* __restrict__ h, int N) {
    int idx = blockIdx.x * blockDim.x + threadIdx.x;   // over N*32 float4s
    if (idx >= N * 32) return;
    int n = idx >> 5, c4 = idx & 31;
    int a = x[n];
    ((v4f*)h)[idx] = ((const v4f*)et)[(size_t)a * 32 + c4];
}

// ---------------------------------------------------------------------------
// WMMA FP32 GEMM:  out[n, colBase:colBase+64] = (h @ W)[n, ...] * ns[n]
// Block: 256 threads = 8 waves; each wave does 16 rows x 64 cols.
// W tile (128 x 64) staged in LDS with row stride 72 floats (bank-friendly).
// Uses V_WMMA_F32_16X16X4_F32 (full fp32 precision; GEMM is not the bottleneck).
// ---------------------------------------------------------------------------
__global__ __launch_bounds__(256) void gemm_ns_kernel(
    const float* __restrict__ h, const float* __restrict__ W,
    const float* __restrict__ ns, float* __restrict__ out, int N)
{
    __shared__ float ldsW[128 * 72];   // 36 KB
    const int tid = threadIdx.x;
    const int colBase = blockIdx.y * 64;

    // Cooperative load of W[0:128][colBase:colBase+64]: 2048 float4s / 256 thr
#pragma unroll
    for (int i = 0; i < 8; ++i) {
        int q   = tid + i * 256;       // float4 index
        int row = q >> 4;              // 16 float4 per row of the 64-col tile
        int c4  = q & 15;
        v4f w = *(const v4f*)(W + (size_t)row * DIM + colBase + c4 * 4);
        *(v4f*)(ldsW + row * 72 + c4 * 4) = w;
    }
    __syncthreads();

    const int lane = tid & 31;
    const int wave = tid >> 5;
    const int m    = lane & 15;        // row within A-tile / col within B-tile
    const int half = lane >> 4;        // K split: lanes 0-15 -> K+{0,1}, 16-31 -> K+{2,3}
    const int r0   = blockIdx.x * 128 + wave * 16;

    const float* hrow = h + (size_t)min(r0 + m, N - 1) * DIM + half * 2;

    v8f acc[4] = {};                   // 4 output tiles of 16x16

    for (int kb = 0; kb < 32; ++kb) {  // K = 128 in steps of 4
        v2f a;
        a.x = hrow[kb * 4 + 0];
        a.y = hrow[kb * 4 + 1];
        const float* wk = ldsW + (kb * 4 + half * 2) * 72 + m;
#pragma unroll
        for (int j = 0; j < 4; ++j) {
            v2f b;
            b.x = wk[j * 16];          // row K+offset
            b.y = wk[j * 16 + 72];     // row K+offset+1
            acc[j] = __builtin_amdgcn_wmma_f32_16x16x4_f32(
                false, a, false, b, (short)0, acc[j], false, false);
        }
    }

#pragma unroll
    for (int r = 0; r < 8; ++r) {
        int row = r0 + r + half * 8;
        if (row < N) {
            float s = ns[row];
            float* orow = out + (size_t)row * DIM + colBase + m;
#pragma unroll
            for (int j = 0; j < 4; ++j) orow[j * 16] = acc[j][r] * s;
        }
    }
}

// ---------------------------------------------------------------------------
// agg[dst[e]] += hw[src[e]]   (32 lanes per edge, float4 chunks, f32 atomics)
// ---------------------------------------------------------------------------
__global__ void scatter_kernel(const float* __restrict__ hw, const int* __restrict__ src,
                               const int* __restrict__ dst, float* __restrict__ agg, int E) {
    int idx = blockIdx.x * blockDim.x + threadIdx.x;   // over E*32
    if (idx >= E * 32) return;
    int e = idx >> 5, c4 = idx & 31;
    int s = src[e], d = dst[e];
    v4f v = ((const v4f*)hw)[(size_t)s * 32 + c4];
    float* p = agg + (size_t)d * DIM + c4 * 4;
    atomicAdd(p + 0, v.x);
    atomicAdd(p + 1, v.y);
    atomicAdd(p + 2, v.z);
    atomicAdd(p + 3, v.w);
}

// ---------------------------------------------------------------------------
// h = (agg*nd + bias) * snorm (in place) + column sum/sumsq for BatchNorm.
// Block handles 64 rows; thread = (rowgrp 0..7, colchunk 0..31).
// ---------------------------------------------------------------------------
__global__ __launch_bounds__(256) void post_stats_kernel(
    float* __restrict__ h, const float* __restrict__ nd, const float* __restrict__ bias,
    const float* __restrict__ sn, float* __restrict__ colsum, float* __restrict__ colsq, int N)
{
    __shared__ v4f ssum[256];
    __shared__ v4f ssq[256];
    const int tid = threadIdx.x;
    const int c4 = tid & 31, rg = tid >> 5;
    const int base = blockIdx.x * 64;
    v4f b4 = ((const v4f*)bias)[c4];
    v4f sum = {}, sq = {};
#pragma unroll
    for (int it = 0; it < 8; ++it) {
        int n = base + rg + it * 8;
        if (n < N) {
            float ndv = nd[n], snv = sn[n];
            v4f v = ((v4f*)h)[(size_t)n * 32 + c4];
            v = (v * ndv + b4) * snv;
            ((v4f*)h)[(size_t)n * 32 + c4] = v;
            sum += v;
            sq  += v * v;
        }
    }
    ssum[tid] = sum; ssq[tid] = sq;
    __syncthreads();
    for (int off = 128; off >= 32; off >>= 1) {
        if (tid < off) { ssum[tid] += ssum[tid + off]; ssq[tid] += ssq[tid + off]; }
        __syncthreads();
    }
    if (tid < 32) {
        v4f s = ssum[tid], q = ssq[tid];
        float* cs = colsum + tid * 4;
        float* cq = colsq  + tid * 4;
        atomicAdd(cs + 0, s.x); atomicAdd(cs + 1, s.y); atomicAdd(cs + 2, s.z); atomicAdd(cs + 3, s.w);
        atomicAdd(cq + 0, q.x); atomicAdd(cq + 1, q.y); atomicAdd(cq + 2, q.z); atomicAdd(cq + 3, q.w);
    }
}

__global__ void bnfinal_kernel(const float* __restrict__ colsum, const float* __restrict__ colsq,
                               const float* __restrict__ gamma, const float* __restrict__ beta,
                               float* __restrict__ scale, float* __restrict__ shift, float invN) {
    int d = threadIdx.x;               // 128 threads
    float mean = colsum[d] * invN;
    float var  = colsq[d] * invN - mean * mean;
    float inv  = rsqrtf(var + 1e-5f);
    float sc   = gamma[d] * inv;
    scale[d] = sc;
    shift[d] = beta[d] - mean * sc;
}

__global__ void bnapply_kernel(float* __restrict__ h, const float* __restrict__ scale,
                               const float* __restrict__ shift, int N) {
    int idx = blockIdx.x * blockDim.x + threadIdx.x;   // over N*32
    if (idx >= N * 32) return;
    int c4 = idx & 31;
    v4f sc = ((const v4f*)scale)[c4];
    v4f sh = ((const v4f*)shift)[c4];
    v4f v = ((v4f*)h)[idx];
    v = v * sc + sh;
    v.x = tanhf(v.x); v.y = tanhf(v.y); v.z = tanhf(v.z); v.w = tanhf(v.w);
    ((v4f*)h)[idx] = v;
}

// ---------------------------------------------------------------------------
// Graph mean pooling (sum + count via atomics)
// ---------------------------------------------------------------------------
__global__ void pool_kernel(const float* __restrict__ h, const int* __restrict__ gid,
                            float* __restrict__ hg, float* __restrict__ cnt, int N) {
    int idx = blockIdx.x * blockDim.x + threadIdx.x;   // over N*32
    if (idx >= N * 32) return;
    int n = idx >> 5, c4 = idx & 31;
    int g = gid[n];
    v4f v = ((const v4f*)h)[idx];
    float* p = hg + (size_t)g * DIM + c4 * 4;
    atomicAdd(p + 0, v.x);
    atomicAdd(p + 1, v.y);
    atomicAdd(p + 2, v.z);
    atomicAdd(p + 3, v.w);
    if (c4 == 0) atomicAdd(cnt + g, 1.0f);
}

// hid[g][j] = relu( sum_d relu(hg[g][d]/cnt[g]) * r1_w[j][d] + r1_b[j] )
__global__ void head1_kernel(const float* __restrict__ hg, const float* __restrict__ cnt,
                             const float* __restrict__ w, const float* __restrict__ b,
                             float* __restrict__ hid) {
    int g = blockIdx.x;
    int j = threadIdx.x;               // 64 threads
    float invc = 1.0f / fmaxf(cnt[g], 1.0f);
    const float* hr = hg + (size_t)g * DIM;
    const float* wr = w + (size_t)j * DIM;
    float acc = 0.0f;
    for (int d = 0; d < DIM; ++d) acc += fmaxf(hr[d] * invc, 0.0f) * wr[d];
    hid[g * 64 + j] = fmaxf(acc + b[j], 0.0f);
}

// out[g] = sum_j hid[g][j]*r2_w[j] + r2_b
__global__ void head2_kernel(const float* __restrict__ hid, const float* __restrict__ w,
                             const float* __restrict__ b, float* __restrict__ out) {
    int g = threadIdx.x;               // G threads
    float acc = 0.0f;
    for (int j = 0; j < 64; ++j) acc += hid[g * 64 + j] * w[j];
    out[g] = acc + b[0];
}

// ---------------------------------------------------------------------------
extern "C" void kernel_launch(void* const* d_in, const int* in_sizes, int n_in,
                              void* d_out, int out_size, void* d_ws, size_t ws_size,
                              hipStream_t stream) {
    const int*   x       = (const int*)d_in[0];
    const int*   src     = (const int*)d_in[2];
    const int*   dst     = (const int*)d_in[3];
    const int*   gid     = (const int*)d_in[4];
    const float* snorm_n = (const float*)d_in[5];
    const float* embed   = (const float*)d_in[7];
    const float* Ws      = (const float*)d_in[8];
    const float* bs      = (const float*)d_in[9];
    const float* gammas  = (const float*)d_in[10];
    const float* betas   = (const float*)d_in[11];
    const float* r1w     = (const float*)d_in[12];
    const float* r1b     = (const float*)d_in[13];
    const float* r2w     = (const float*)d_in[14];
    const float* r2b     = (const float*)d_in[15];
    float* out = (float*)d_out;

    const int N = in_sizes[0];
    const int E = in_sizes[2];
    const int G = out_size;                      // 128
    const int L = in_sizes[8] / (DIM * DIM);     // 8

    // Workspace carve-out (256B aligned slices)
    char* wp = (char*)d_ws;
    auto take = [&](size_t bytes) -> void* {
        void* p = (void*)wp;
        wp += (bytes + 255) & ~(size_t)255;
        return p;
    };
    float* h      = (float*)take((size_t)N * DIM * 4);
    float* hw     = (float*)take((size_t)N * DIM * 4);
    float* ns     = (float*)take((size_t)N * 4);
    float* nd     = (float*)take((size_t)N * 4);
    float* colsum = (float*)take(512);
    float* colsq  = (float*)take(512);
    float* scale  = (float*)take(512);
    float* shift  = (float*)take(512);
    float* hg     = (float*)take((size_t)G * DIM * 4);
    float* cnt    = (float*)take((size_t)G * 4);
    float* hid    = (float*)take((size_t)G * 64 * 4);
    (void)ws_size; (void)n_in;

    // Degrees -> ns/nd
    hipMemsetAsync(ns, 0, (size_t)N * 4, stream);
    hipMemsetAsync(nd, 0, (size_t)N * 4, stream);
    degree_kernel<<<(E + 255) / 256, 256, 0, stream>>>(src, dst, ns, nd, E);
    invsqrt_kernel<<<(N + 255) / 256, 256, 0, stream>>>(ns, nd, N);

    // Embedding lookup
    embed_kernel<<<(N * 32 + 255) / 256, 256, 0, stream>>>(x, embed, h, N);

    for (int i = 0; i < L; ++i) {
        dim3 gg((N + 127) / 128, 2);
        gemm_ns_kernel<<<gg, 256, 0, stream>>>(h, Ws + (size_t)i * DIM * DIM, ns, hw, N);

        hipMemsetAsync(h, 0, (size_t)N * DIM * 4, stream);   // h becomes agg
        scatter_kernel<<<(E * 32 + 255) / 256, 256, 0, stream>>>(hw, src, dst, h, E);

        hipMemsetAsync(colsum, 0, 512, stream);
        hipMemsetAsync(colsq, 0, 512, stream);
        post_stats_kernel<<<(N + 63) / 64, 256, 0, stream>>>(h, nd, bs + i * DIM, snorm_n,
                                                             colsum, colsq, N);
        bnfinal_kernel<<<1, 128, 0, stream>>>(colsum, colsq, gammas + i * DIM, betas + i * DIM,
                                              scale, shift, 1.0f / (float)N);
        bnapply_kernel<<<(N * 32 + 255) / 256, 256, 0, stream>>>(h, scale, shift, N);
    }

    // Graph pooling + MLP head
    hipMemsetAsync(hg, 0, (size_t)G * DIM * 4, stream);
    hipMemsetAsync(cnt, 0, (size_t)G * 4, stream);
    pool_kernel<<<(N * 32 + 255) / 256, 256, 0, stream>>>(h, gid, hg, cnt, N);
    head1_kernel<<<G, 64, 0, stream>>>(hg, cnt, r1w, r1b, hid);
    head2_kernel<<<1, G, 0, stream>>>(hid, r2w, r2b, out);
}